// CondNetInference_87110526697918
// MI455X (gfx1250) — compile-verified
//
#include <hip/hip_runtime.h>
#include <hip/hip_bf16.h>
#include <stdint.h>

#define N_TOTAL  1048576
#define G_GRAPHS 256
#define NPG      4096          // nodes per graph (contiguous segments)
#define BLK      1024          // 32 wave32 waves
#define NPT      (NPG / BLK)   // 4 nodes per thread
#define T_B      0.5f
#define T_D      1.5f

typedef unsigned int v4u __attribute__((ext_vector_type(4)));
typedef int          v8i __attribute__((ext_vector_type(8)));
typedef int          v4i __attribute__((ext_vector_type(4)));

__device__ __forceinline__ uint64_t pack_key(float v, uint32_t idx) {
  // v >= 0 so float bits are order-preserving; ~idx makes max-key pick min idx on ties
  return ((uint64_t)__float_as_uint(v) << 32) | (uint32_t)(~idx);
}

// ---------------------------------------------------------------------------
// Kernel 1: per-graph greedy condensation-point clustering.
// One block per graph. x tile staged to LDS via the Tensor Data Mover (TDM).
// ---------------------------------------------------------------------------
__global__ __launch_bounds__(BLK) void cond_points_kernel(
    const float* __restrict__ x, const float* __restrict__ beta,
    float* __restrict__ out_iscp, float* __restrict__ out_np) {
  __shared__ float    s_x[NPG * 3];   // 48 KB: this graph's x tile
  __shared__ uint64_t s_red[32];
  __shared__ uint64_t s_key;

  const int    g    = blockIdx.x;
  const int    tid  = threadIdx.x;
  const int    lane = tid & 31;
  const int    wid  = tid >> 5;
  const size_t base = (size_t)g * NPG;

  // --- stage x tile (12288 f32) into LDS -----------------------------------
#if defined(__gfx1250__) && __has_builtin(__builtin_amdgcn_tensor_load_to_lds)
  if (tid < 32) {                       // single wave issues the DMA
    uint64_t ga    = (uint64_t)(uintptr_t)(x + base * 3);
    uint32_t ldsa  = (uint32_t)(uintptr_t)(&s_x[0]);
    const uint32_t nelem = NPG * 3;     // 12288 elements of 4 bytes
    // D# group 0: count=1 | lds_addr | global_addr[56:0] | type=2
    v4u g0;
    g0.x = 1u;
    g0.y = ldsa;
    g0.z = (uint32_t)ga;
    g0.w = (uint32_t)((ga >> 32) & 0x1FFFFFFu) | (2u << 30);
    // D# group 1: data_size=4B, 1-D tile: tile_dim0=nelem, tensor_dim0=nelem
    v8i g1;
    g1[0] = (int)(2u << 16);                                  // data_size=2 (4B)
    g1[1] = (int)((nelem & 0xFFFFu) << 16);                   // tensor_dim0 lo
    g1[2] = (int)(((nelem >> 16) & 0xFFFFu) | (1u << 16));    // td0 hi | tensor_dim1=1
    g1[3] = (int)((nelem & 0xFFFFu) << 16);                   // tile_dim0
    g1[4] = 0;                                                // tile_dim1/2 unused
    g1[5] = (int)nelem;                                       // tensor_dim0_stride lo
    g1[6] = 0;
    g1[7] = 0;
    v4i gz4 = {0, 0, 0, 0};
    v8i gz8 = {0, 0, 0, 0, 0, 0, 0, 0};
    // 6-arg form (this toolchain): (g0, g1, g2, g3, g4, cpol)
    __builtin_amdgcn_tensor_load_to_lds(g0, g1, gz4, gz4, gz8, 0);
#if __has_builtin(__builtin_amdgcn_s_wait_tensorcnt)
    __builtin_amdgcn_s_wait_tensorcnt(0);
#endif
  }
#else
  for (int i = tid; i < NPG * 3; i += BLK) s_x[i] = x[base * 3 + i];
#endif
  __syncthreads();

  // --- per-thread node state in registers ----------------------------------
  float mb[NPT];     // mailbox beta: beta * (1 - assigned)
  float iscp[NPT];
#pragma unroll
  for (int k = 0; k < NPT; ++k) {
    mb[k]   = beta[base + tid + k * BLK];
    iscp[k] = 0.0f;
  }

  int count = 0;
  for (;;) {
    // local best over my 4 nodes
    uint64_t best = 0;
#pragma unroll
    for (int k = 0; k < NPT; ++k) {
      uint64_t kk = pack_key(mb[k], (uint32_t)(tid + k * BLK));
      best = (kk > best) ? kk : best;
    }
    // wave32 reduction
#pragma unroll
    for (int off = 16; off > 0; off >>= 1) {
      uint64_t o = __shfl_xor((unsigned long long)best, off, 32);
      if (o > best) best = o;
    }
    if (lane == 0) s_red[wid] = best;
    __syncthreads();
    // cross-wave reduction (wave 0 reduces 32 partials)
    if (wid == 0) {
      uint64_t b = s_red[lane];
#pragma unroll
      for (int off = 16; off > 0; off >>= 1) {
        uint64_t o = __shfl_xor((unsigned long long)b, off, 32);
        if (o > b) b = o;
      }
      if (lane == 0) s_key = b;
    }
    __syncthreads();

    const uint64_t kfin = s_key;
    const float    vmax = __uint_as_float((uint32_t)(kfin >> 32));
    if (vmax < T_B) break;                     // uniform exit
    const uint32_t idx = ~((uint32_t)kfin);    // local node index 0..4095

    // mark condensation point (owner thread only)
    if ((int)(idx & (BLK - 1)) == tid) iscp[idx / BLK] = 1.0f;
    ++count;                                   // uniform across block

    // broadcast selected point's x from LDS, assign nodes within T_D
    const float bx0 = s_x[idx * 3 + 0];
    const float bx1 = s_x[idx * 3 + 1];
    const float bx2 = s_x[idx * 3 + 2];
#pragma unroll
    for (int k = 0; k < NPT; ++k) {
      const int node = tid + k * BLK;
      const float dx = s_x[node * 3 + 0] - bx0;
      const float dy = s_x[node * 3 + 1] - bx1;
      const float dz = s_x[node * 3 + 2] - bx2;
      if (dx * dx + dy * dy + dz * dz <= T_D * T_D) mb[k] = 0.0f;
    }
  }

#pragma unroll
  for (int k = 0; k < NPT; ++k)
    out_iscp[base + tid + k * BLK] = iscp[k];
  if (tid == 0) out_np[g] = (float)count;
}

// ---------------------------------------------------------------------------
// Kernel 2: elementwise undo-scaling + argmax + masked writes (bandwidth-bound)
// ---------------------------------------------------------------------------
__global__ __launch_bounds__(256) void finalize_kernel(
    const float* __restrict__ pt4, const float* __restrict__ pos,
    const float* __restrict__ clsp, const float* __restrict__ chgp,
    const float* __restrict__ iscp, float* __restrict__ out) {
  const int i = blockIdx.x * blockDim.x + threadIdx.x;
  if (i >= N_TOTAL) return;

  const float m = iscp[i];

  const float a0 = pt4[i * 4 + 0];
  const float a1 = pt4[i * 4 + 1];
  const float a2 = pt4[i * 4 + 2];
  const float a3 = pt4[i * 4 + 3];

  const float pt    = expf(1.0f * a0 + 4.0f);     // PT_STD, PT_MEAN
  const float eta   = 2.0f * a1;                  // ETA_STD, ETA_MEAN
  const float xhat  = 0.7f * a2;
  const float yhat  = 0.7f * a3;
  const float phi   = atan2f(yhat, xhat);
  const float px    = pt * cosf(phi);
  const float py    = pt * sinf(phi);
  const float theta = 2.0f * atanf(-expf(eta));
  const float pz    = pt / tanf(theta);

  // argmax over 5 classes (first-occurrence on ties)
  int   ci = 0;
  float cv = clsp[i * 5 + 0];
#pragma unroll
  for (int c = 1; c < 5; ++c) {
    const float v = clsp[i * 5 + c];
    if (v > cv) { cv = v; ci = c; }
  }
  // argmax over 3 charges
  int   qi = 0;
  float qv = chgp[i * 3 + 0];
#pragma unroll
  for (int c = 1; c < 3; ++c) {
    const float v = chgp[i * 3 + c];
    if (v > qv) { qv = v; qi = c; }
  }

  const size_t Nz = (size_t)N_TOTAL;
  out[0 * Nz + i] = eta * m;
  out[1 * Nz + i] = phi * m;
  out[2 * Nz + i] = pt * m;
  out[3 * Nz + (size_t)i * 3 + 0] = px * m;
  out[3 * Nz + (size_t)i * 3 + 1] = py * m;
  out[3 * Nz + (size_t)i * 3 + 2] = pz * m;
  out[6 * Nz + (size_t)i * 3 + 0] = pos[i * 3 + 0] * 30.0f * m;
  out[6 * Nz + (size_t)i * 3 + 1] = pos[i * 3 + 1] * 30.0f * m;
  out[6 * Nz + (size_t)i * 3 + 2] = pos[i * 3 + 2] * 30.0f * m;
  out[9 * Nz + i]  = (float)ci * m;
  out[10 * Nz + i] = (float)qi * m;
  // out[11N..12N) = is_cp and out[12N..12N+G) = num_particles written by kernel 1
}

extern "C" void kernel_launch(void* const* d_in, const int* in_sizes, int n_in,
                              void* d_out, int out_size, void* d_ws, size_t ws_size,
                              hipStream_t stream) {
  (void)in_sizes; (void)n_in; (void)out_size; (void)d_ws; (void)ws_size;
  const float* x    = (const float*)d_in[0];
  const float* beta = (const float*)d_in[1];
  const float* pt4  = (const float*)d_in[2];
  const float* pos  = (const float*)d_in[3];
  const float* clsp = (const float*)d_in[4];
  const float* chgp = (const float*)d_in[5];
  // d_in[6] node_graph_idx: contiguous equal segments by construction
  float* out = (float*)d_out;
  float* out_iscp = out + (size_t)11 * N_TOTAL;
  float* out_np   = out + (size_t)12 * N_TOTAL;

  cond_points_kernel<<<G_GRAPHS, BLK, 0, stream>>>(x, beta, out_iscp, out_np);
  finalize_kernel<<<N_TOTAL / 256, 256, 0, stream>>>(pt4, pos, clsp, chgp, out_iscp, out);
}